// gPool_88364657147965
// MI455X (gfx1250) — compile-verified
//
#include <hip/hip_runtime.h>
#include <hip/hip_bf16.h>

#define N_NODES 16384
#define FEAT    256
#define K_SEL   8192

typedef float v2f __attribute__((ext_vector_type(2)));
typedef float v4f __attribute__((ext_vector_type(4)));
typedef float v8f __attribute__((ext_vector_type(8)));

// ---------------------------------------------------------------------------
// K1: score_raw = x @ W via V_WMMA_F32_16X16X4_F32.
// One wave -> 16 rows. A tile (16x4 f32): lane layout M = lane%16,
// K = vgpr + 2*(lane>=16)  => per-lane contiguous float2 of the x row.
// B tile (4x16): same K mapping; W replicated across all 16 columns so every
// column of D equals the 16 row-scores (robust to N-column mapping).
// ---------------------------------------------------------------------------
__global__ void __launch_bounds__(128)
k_gemv_wmma(const float* __restrict__ x, const float* __restrict__ w,
            float* __restrict__ raw) {
    const int lane  = threadIdx.x & 31;
    const int wave  = threadIdx.x >> 5;
    const int half  = lane >> 4;         // 0: K pair {0,1}, 1: K pair {2,3}
    const int mrow  = lane & 15;         // M for the A operand
    const int mbase = (blockIdx.x * 4 + wave) * 16;

    const float* __restrict__ xrow = x + (size_t)(mbase + mrow) * FEAT;

    v8f c = {0.f, 0.f, 0.f, 0.f, 0.f, 0.f, 0.f, 0.f};

    for (int k = 0; k < FEAT; k += 4) {
        const int kk = k + 2 * half;
        v2f a; a.x = xrow[kk]; a.y = xrow[kk + 1];
        v2f b; b.x = w[kk];    b.y = w[kk + 1];
        c = __builtin_amdgcn_wmma_f32_16x16x4_f32(
                /*neg_a=*/false, a, /*neg_b=*/false, b,
                /*c_mod=*/(short)0, c, /*reuse_a=*/false, /*reuse_b=*/false);
    }

    // D layout: c[v] = D[v + 8*half, lane%16]; all columns identical.
    if ((lane & 15) == 0) {
        const int rbase = mbase + 8 * half;
        #pragma unroll
        for (int v = 0; v < 8; ++v) raw[rbase + v] = c[v];
    }
}

// ---------------------------------------------------------------------------
// K2: single-block global L2 normalization + tanh.
// ---------------------------------------------------------------------------
__global__ void __launch_bounds__(256)
k_norm_tanh(const float* __restrict__ raw, float* __restrict__ score) {
    __shared__ float red[256];
    float s = 0.f;
    for (int i = threadIdx.x; i < N_NODES; i += 256) {
        float v = raw[i];
        s += v * v;
    }
    red[threadIdx.x] = s;
    __syncthreads();
    for (int off = 128; off > 0; off >>= 1) {
        if ((int)threadIdx.x < off) red[threadIdx.x] += red[threadIdx.x + off];
        __syncthreads();
    }
    const float inv = 1.0f / sqrtf(red[0]);
    for (int i = threadIdx.x; i < N_NODES; i += 256) {
        score[i] = tanhf(raw[i] * inv);
    }
}

// ---------------------------------------------------------------------------
// K3: top-K via stable descending rank (matches jax.lax.top_k ordering):
//   rank_i = #{j : s_j > s_i} + #{j < i : s_j == s_i}
// Element i with rank < K lands at slot `rank`. Ranks of selected elements
// form a permutation of [0,K) so every slot is written exactly once.
// Also emits idx (as float) into the third output segment.
// ---------------------------------------------------------------------------
__global__ void __launch_bounds__(256)
k_rank_topk(const float* __restrict__ score, int* __restrict__ idx_out,
            float* __restrict__ val_out, float* __restrict__ idx_f_out) {
    __shared__ float tile[256];
    const int i  = blockIdx.x * 256 + threadIdx.x;
    const float si = score[i];
    int rank = 0;
    for (int base = 0; base < N_NODES; base += 256) {
        __syncthreads();
        tile[threadIdx.x] = score[base + threadIdx.x];
        __syncthreads();
        #pragma unroll 8
        for (int t = 0; t < 256; ++t) {
            const float sj = tile[t];
            const int   j  = base + t;
            rank += (sj > si) | ((sj == si) & (j < i));
        }
    }
    if (rank < K_SEL) {
        idx_out[rank]   = i;
        val_out[rank]   = si;
        idx_f_out[rank] = (float)i;
    }
}

// ---------------------------------------------------------------------------
// K4: new_x[r, :] = x[idx[r], :] * values[r]
// 128-bit vector per lane (64 lanes/row); NT stores: output is never re-read.
// ---------------------------------------------------------------------------
__global__ void __launch_bounds__(64)
k_new_x(const float* __restrict__ x, const int* __restrict__ idx,
        const float* __restrict__ vals, float* __restrict__ out) {
    const int r   = blockIdx.x;
    const int src = idx[r];
    const float v = vals[r];
    const v4f* __restrict__ xin =
        (const v4f*)(x + (size_t)src * FEAT) + threadIdx.x;
    v4f* __restrict__ op =
        (v4f*)(out + (size_t)r * FEAT) + threadIdx.x;
    v4f t = *xin;
    t *= v;
    __builtin_nontemporal_store(t, op);
}

// ---------------------------------------------------------------------------
// K5: new_A[r, c] = A[idx[r], idx[c]]
// One workgroup per output row: the 64KB source row stays hot in L2 while
// its 8192 random column gathers replay against it (each 128B line hit ~16x).
// Writes are coalesced and non-temporal (256 MB streamed, never re-read) so
// they don't evict the read-side working set from L2.
// ---------------------------------------------------------------------------
__global__ void __launch_bounds__(256)
k_new_A(const float* __restrict__ A, const int* __restrict__ idx,
        float* __restrict__ out) {
    const int r = blockIdx.x;
    const float* __restrict__ arow = A + (size_t)idx[r] * N_NODES;
    float* __restrict__ orow = out + (size_t)r * K_SEL;
    #pragma unroll 4
    for (int c = threadIdx.x; c < K_SEL; c += 256) {
        __builtin_nontemporal_store(arow[idx[c]], orow + c);
    }
}

// ---------------------------------------------------------------------------
extern "C" void kernel_launch(void* const* d_in, const int* in_sizes, int n_in,
                              void* d_out, int out_size, void* d_ws, size_t ws_size,
                              hipStream_t stream) {
    (void)in_sizes; (void)n_in; (void)out_size; (void)ws_size;

    const float* x = (const float*)d_in[0];   // [N, FEAT]
    const float* A = (const float*)d_in[1];   // [N, N]
    const float* W = (const float*)d_in[2];   // [FEAT]

    // Workspace layout (bytes): raw[N] | score[N] | idx[K] | vals[K]
    char* ws = (char*)d_ws;
    float* raw   = (float*)(ws);
    float* score = (float*)(ws + (size_t)N_NODES * 4);
    int*   idx   = (int*)  (ws + (size_t)N_NODES * 8);
    float* vals  = (float*)(ws + (size_t)N_NODES * 8 + (size_t)K_SEL * 4);

    // Output layout (floats): new_x [K*FEAT] | new_A [K*K] | idx [K]
    float* out     = (float*)d_out;
    float* out_x   = out;
    float* out_A   = out + (size_t)K_SEL * FEAT;
    float* out_idx = out + (size_t)K_SEL * FEAT + (size_t)K_SEL * K_SEL;

    // K1: raw scores via WMMA GEMV (1024 waves, 4 waves/block)
    k_gemv_wmma<<<N_NODES / 64, 128, 0, stream>>>(x, W, raw);

    // K2: global L2-normalize + tanh
    k_norm_tanh<<<1, 256, 0, stream>>>(raw, score);

    // K3: stable rank top-K (also writes idx-as-float output)
    k_rank_topk<<<N_NODES / 256, 256, 0, stream>>>(score, idx, vals, out_idx);

    // K4: gather + scale selected feature rows (vector loads, NT stores)
    k_new_x<<<K_SEL, 64, 0, stream>>>(x, idx, vals, out_x);

    // K5: subgraph adjacency gather (NT stores)
    k_new_A<<<K_SEL, 256, 0, stream>>>(A, idx, out_A);
}